// MultiS2V_89996744720444
// MI455X (gfx1250) — compile-verified
//
#include <hip/hip_runtime.h>
#include <stdint.h>

// ---------------------------------------------------------------------------
// MultiS2V heterogeneous-GNN forward for MI455X (gfx1250, wave32).
//
// Roofline: scatter/segment_sum dominates (~3.9 GB of gather+atomic traffic
// ~= 170us @ 23.3 TB/s; the ~84 MB acc working set is resident in the 192 MB
// L2, so the fp32 atomics resolve at L2). All GEMMs (~12 GFLOP fp32) run on
// V_WMMA_F32_16X16X4_F32 so they stay memory-bound at full fp32 precision.
// ---------------------------------------------------------------------------

typedef float v2f __attribute__((ext_vector_type(2)));
typedef float v8f __attribute__((ext_vector_type(8)));

// Hardware fp32 atomic add, no return (GLOBAL_ATOMIC_ADD_F32, STOREcnt).
// Inline asm guarantees the L2 atomic path (never a CAS loop). S_ENDPGM does
// an implicit wait-idle, and kernel boundaries order it vs later kernels.
__device__ __forceinline__ void atomAddF32(float* p, float v) {
#if defined(__HIP_DEVICE_COMPILE__)
    uint64_t ap = (uint64_t)p;
    asm volatile("global_atomic_add_f32 %0, %1, off"
                 :
                 : "v"(ap), "v"(v)
                 : "memory");
#else
    (void)p; (void)v;
#endif
}

// ---------------------------------------------------------------------------
// Generic GEMM:  Out[N, ldo] = epilogue( X[N,K] @ W[:, :K]^T )
//   W is row-major [M, ldw] (we use its first K columns -> supports using the
//   right half of w5 via pointer offset + ldw=2C).
//   epilogue: v = alpha * (acc + bias[m]);  if relu: v = max(v, 0)
// One wave computes one 16x16 output tile; 8 waves / block.
// A-frag: lane(m=lane&15, kp=(lane>>4)*2) = X[row0+m][k+kp .. k+kp+1]  (float2)
// B-frag: lane(n=lane&15, kp)             = W[col0+n][k+kp .. k+kp+1]  (float2)
// D: vgpr r -> Out[row0 + r + 8*(lane>>4)][col0 + (lane&15)]
// ---------------------------------------------------------------------------
__global__ void __launch_bounds__(256) gemm_wmma_f32(
    const float* __restrict__ X,
    const float* __restrict__ W, int ldw,
    const float* __restrict__ bias,     // may be nullptr
    float* __restrict__ Out, int ldo,
    int N, int K, int M,
    float alpha, int relu)
{
    const int lane   = threadIdx.x & 31;
    const int wave   = threadIdx.x >> 5;
    const int tilesM = M >> 4;
    const int tiles  = (N >> 4) * tilesM;
    const int tile   = blockIdx.x * 8 + wave;
    if (tile >= tiles) return;          // wave-uniform exit: EXEC all-ones for WMMA

    const int tm   = tile % tilesM;
    const int tn   = tile / tilesM;
    const int row0 = tn << 4;
    const int col0 = tm << 4;
    const int mn   = lane & 15;
    const int kp   = (lane >> 4) << 1;

    const float* xp = X + (size_t)(row0 + mn) * K   + kp;
    const float* wp = W + (size_t)(col0 + mn) * ldw + kp;

    v8f c = {};
    for (int k = 0; k < K; k += 4) {
        v2f a = *reinterpret_cast<const v2f*>(xp + k);
        v2f b = *reinterpret_cast<const v2f*>(wp + k);
        c = __builtin_amdgcn_wmma_f32_16x16x4_f32(
                /*neg_a=*/false, a, /*neg_b=*/false, b,
                /*c_mod=*/(short)0, c, /*reuse_a=*/false, /*reuse_b=*/false);
    }

    const float bv = bias ? bias[col0 + mn] : 0.0f;
    float* op = Out + (size_t)(row0 + ((lane >> 4) << 3)) * ldo + col0 + mn;
#pragma unroll
    for (int r = 0; r < 8; ++r) {
        float v = alpha * (c[r] + bv);
        if (relu) v = fmaxf(v, 0.0f);
        op[(size_t)r * ldo] = v;
    }
}

// ---------------------------------------------------------------------------
// Edge scatter:  Acc[dst] += Y[src]   (rows of length C, float4 granularity)
// edges layout: [2, E] int32, row0 = src, row1 = dst.
// ---------------------------------------------------------------------------
__global__ void __launch_bounds__(256) scatter_add_kernel(
    const int* __restrict__ edges, int E,
    const float* __restrict__ Y,
    float* __restrict__ Acc,
    int C, int lc)                       // lc = log2(C/4)
{
    const long long t = (long long)blockIdx.x * blockDim.x + threadIdx.x;
    if (t >= ((long long)E << lc)) return;
    const int e  = (int)(t >> lc);
    const int ch = (((int)t) & ((1 << lc) - 1)) << 2;
    const int src = edges[e];
    const int dst = edges[E + e];
    const float4 v = *reinterpret_cast<const float4*>(Y + (size_t)src * C + ch);
    float* ap = Acc + (size_t)dst * C + ch;
    atomAddF32(ap + 0, v.x);
    atomAddF32(ap + 1, v.y);
    atomAddF32(ap + 2, v.z);
    atomAddF32(ap + 3, v.w);
}

// ---------------------------------------------------------------------------
// Column sum:  out[c] += sum_{rows in slab} A[r][c].  blockDim = C.
// ---------------------------------------------------------------------------
__global__ void colsum_kernel(const float* __restrict__ A, int N, int C,
                              float* __restrict__ out)
{
    const int c  = threadIdx.x;
    const int r0 = blockIdx.x * 512;
    int r1 = r0 + 512; if (r1 > N) r1 = N;
    float s = 0.0f;
    for (int r = r0; r < r1; ++r) s += A[(size_t)r * C + c];
    atomAddF32(&out[c], s);
}

__global__ void zero_f32(float* p, int n) {
    int t = blockIdx.x * blockDim.x + threadIdx.x;
    if (t < n) p[t] = 0.0f;
}

// biascomb_i = b1_i + b2[0][i] + b2[1][i] + b2[2][i]
__global__ void bias4_kernel(const float* a, const float* b, const float* c,
                             const float* d, float* out, int n) {
    int t = threadIdx.x;
    if (t < n) out[t] = a[t] + b[t] + c[t] + d[t];
}

// left_i[c] = sum_j (sums_j . w6[j][i][c,:]) + b6[j][i][c]
// const_i[o] = b5[o] + sum_c w5[o][c] * relu(left_i[c])
__global__ void leftconst_kernel(
    const float* __restrict__ s0, const float* __restrict__ s1, const float* __restrict__ s2,
    const float* __restrict__ w6_0, const float* __restrict__ w6_1, const float* __restrict__ w6_2,
    const float* __restrict__ b6_0, const float* __restrict__ b6_1, const float* __restrict__ b6_2,
    int C0, int C1, int C2, int Ci,
    const float* __restrict__ w5, const float* __restrict__ b5, int OUTi,
    float* __restrict__ constb)
{
    __shared__ float lds[64];
    const int t = threadIdx.x;
    if (t < Ci) {
        float acc = b6_0[t] + b6_1[t] + b6_2[t];
        for (int k = 0; k < C0; ++k) acc += s0[k] * w6_0[t * C0 + k];
        for (int k = 0; k < C1; ++k) acc += s1[k] * w6_1[t * C1 + k];
        for (int k = 0; k < C2; ++k) acc += s2[k] * w6_2[t * C2 + k];
        lds[t] = fmaxf(acc, 0.0f);
    }
    __syncthreads();
    if (t < OUTi) {
        float acc = b5[t];
        const float* wrow = w5 + (size_t)t * (2 * Ci);   // left half columns
        for (int c = 0; c < Ci; ++c) acc += wrow[c] * lds[c];
        constb[t] = acc;
    }
}

// ---------------------------------------------------------------------------
extern "C" void kernel_launch(void* const* d_in, const int* in_sizes, int n_in,
                              void* d_out, int out_size, void* d_ws, size_t ws_size,
                              hipStream_t stream)
{
    (void)n_in; (void)out_size; (void)ws_size;
    const int C[3]    = {64, 64, 32};
    const int OUTC[3] = {64, 64, 32};

    const float* x[3];
    int N[3];
    for (int i = 0; i < 3; ++i) {
        x[i] = (const float*)d_in[i];
        N[i] = in_sizes[i] / C[i];
    }
    const int* e[3][3];
    for (int j = 0; j < 3; ++j)
        for (int i = 0; i < 3; ++i)
            e[j][i] = (const int*)d_in[3 + j * 3 + i];
    const int E = in_sizes[3] / 2;

    // params: pytree flatten of dict, keys sorted: b1,b2,b5,b6,b7,w1,w2,w5,w6,w7
    int p = 12;
    const float *b1[3], *b2[3][3], *b5[3], *b6[3][3], *b7[3];
    const float *w1[3], *w2[3][3], *w5[3], *w6[3][3], *w7[3];
    for (int i = 0; i < 3; ++i) b1[i] = (const float*)d_in[p++];
    for (int j = 0; j < 3; ++j) for (int i = 0; i < 3; ++i) b2[j][i] = (const float*)d_in[p++];
    for (int i = 0; i < 3; ++i) b5[i] = (const float*)d_in[p++];
    for (int j = 0; j < 3; ++j) for (int i = 0; i < 3; ++i) b6[j][i] = (const float*)d_in[p++];
    for (int i = 0; i < 3; ++i) b7[i] = (const float*)d_in[p++];
    for (int i = 0; i < 3; ++i) w1[i] = (const float*)d_in[p++];
    for (int j = 0; j < 3; ++j) for (int i = 0; i < 3; ++i) w2[j][i] = (const float*)d_in[p++];
    for (int i = 0; i < 3; ++i) w5[i] = (const float*)d_in[p++];
    for (int j = 0; j < 3; ++j) for (int i = 0; i < 3; ++i) w6[j][i] = (const float*)d_in[p++];
    for (int i = 0; i < 3; ++i) w7[i] = (const float*)d_in[p++];

    // -------- workspace carve (floats) --------
    float* ws = (float*)d_ws;
    size_t off = 0;
    float* acc[3];
    for (int i = 0; i < 3; ++i) { acc[i] = ws + off; off += (size_t)N[i] * C[i]; }
    size_t maxNC = 0;
    for (int j = 0; j < 3; ++j)
        for (int i = 0; i < 3; ++i) {
            size_t s = (size_t)N[j] * C[i];
            if (s > maxNC) maxNC = s;
        }
    float* ybuf = ws + off; off += maxNC;                 // y_ji, later h_i (reused)
    float* sums[3];  for (int i = 0; i < 3; ++i) { sums[i]  = ws + off; off += C[i]; }
    float* bcomb[3]; for (int i = 0; i < 3; ++i) { bcomb[i] = ws + off; off += C[i]; }
    float* cb[3];    for (int i = 0; i < 3; ++i) { cb[i]    = ws + off; off += OUTC[i]; }

    // -------- small prep --------
    zero_f32<<<1, 256, 0, stream>>>(sums[0], C[0] + C[1] + C[2]);  // contiguous
    for (int i = 0; i < 3; ++i)
        bias4_kernel<<<1, C[i], 0, stream>>>(b1[i], b2[0][i], b2[1][i], b2[2][i],
                                             bcomb[i], C[i]);

    // -------- acc_i = x_i @ w1_i^T + (b1_i + sum_j b2[j][i]) --------
    for (int i = 0; i < 3; ++i) {
        const int tiles = (N[i] >> 4) * (C[i] >> 4);
        gemm_wmma_f32<<<(tiles + 7) / 8, 256, 0, stream>>>(
            x[i], w1[i], C[i], bcomb[i], acc[i], C[i],
            N[i], C[i], C[i], 1.0f, 0);
    }

    // -------- message passing: y = x_j @ w2[j][i]^T ; acc_i[dst] += y[src] --------
    for (int i = 0; i < 3; ++i) {
        const int lc = (C[i] == 64) ? 4 : 3;              // log2(C/4)
        for (int j = 0; j < 3; ++j) {
            const int tiles = (N[j] >> 4) * (C[i] >> 4);
            gemm_wmma_f32<<<(tiles + 7) / 8, 256, 0, stream>>>(
                x[j], w2[j][i], C[j], nullptr, ybuf, C[i],
                N[j], C[j], C[i], 1.0f, 0);
            const long long tot = (long long)E << lc;
            scatter_add_kernel<<<(int)((tot + 255) / 256), 256, 0, stream>>>(
                e[j][i], E, ybuf, acc[i], C[i], lc);
        }
    }

    // -------- sums_j = feats_j.sum(axis=0) --------
    for (int i = 0; i < 3; ++i)
        colsum_kernel<<<(N[i] + 511) / 512, C[i], 0, stream>>>(acc[i], N[i], C[i], sums[i]);

    // -------- fold lin6 + left-half of lin5 into per-channel constant --------
    for (int i = 0; i < 3; ++i)
        leftconst_kernel<<<1, 64, 0, stream>>>(
            sums[0], sums[1], sums[2],
            w6[0][i], w6[1][i], w6[2][i],
            b6[0][i], b6[1][i], b6[2][i],
            C[0], C[1], C[2], C[i],
            w5[i], b5[i], OUTC[i], cb[i]);

    // -------- out_i = relu(3*(acc_i @ w7^T + b7)) @ w5[:,C:]^T + const_i --------
    float* outp = (float*)d_out;
    size_t oo = 0;
    for (int i = 0; i < 3; ++i) {
        const int tilesA = (N[i] >> 4) * (C[i] >> 4);
        gemm_wmma_f32<<<(tilesA + 7) / 8, 256, 0, stream>>>(
            acc[i], w7[i], C[i], b7[i], ybuf, C[i],
            N[i], C[i], C[i], 3.0f, 1);
        const int tilesB = (N[i] >> 4) * (OUTC[i] >> 4);
        gemm_wmma_f32<<<(tilesB + 7) / 8, 256, 0, stream>>>(
            ybuf, w5[i] + C[i], 2 * C[i], cb[i], outp + oo, OUTC[i],
            N[i], C[i], OUTC[i], 1.0f, 0);
        oo += (size_t)N[i] * OUTC[i];
    }
}